// ForceGrid_5875515261578
// MI455X (gfx1250) — compile-verified
//
#include <hip/hip_runtime.h>

// ForceGrid scatter: 10M particles -> 256^3 fp32 grid via nearest-grid-point
// deposition with atomic adds.
//
// MI455X strategy:
//  - 64MB grid fits in 192MB L2 -> atomics resolve in L2, no HBM RMW.
//  - Particle stream (160MB, read-once) loaded with non-temporal hint so it
//    does not evict the grid working set from L2.
//  - Fire-and-forget global_atomic_add_f32 (no return -> STOREcnt path).
//  - global_prefetch_b8 one grid-stride ahead to hide stream latency.
//
// scatter_kernel is defined FIRST so the disasm snippet shows the hot kernel.

#define GRID_N    256
#define GRID_EDGE 10.0f

__global__ void __launch_bounds__(256) scatter_kernel(
    const float* __restrict__ pos,   // [N,3] row-major
    const float* __restrict__ wgt,   // [N]
    float* __restrict__ grid,        // [256^3]
    int n) {
  // dx = (max - min)/(n-1) computed in fp32, exactly as the reference.
  const float gmin = -GRID_EDGE;
  const float dx = (2.0f * GRID_EDGE) / (float)(GRID_N - 1);

  const int stride = gridDim.x * blockDim.x;
  for (int i = blockIdx.x * blockDim.x + threadIdx.x; i < n; i += stride) {
    // Prefetch this thread's next-iteration particle row (one loop iteration
    // of latency distance; in-flight prefetch window ~20MB << 192MB L2).
    if (i + stride < n) {
      __builtin_prefetch(&pos[3 * (i + stride)], 0, 0);
    }

    // Streaming, read-once data: non-temporal loads keep the 64MB grid
    // resident in the 192MB L2.
    const int j = 3 * i;
    float px = __builtin_nontemporal_load(&pos[j + 0]);
    float py = __builtin_nontemporal_load(&pos[j + 1]);
    float pz = __builtin_nontemporal_load(&pos[j + 2]);
    float w  = __builtin_nontemporal_load(&wgt[i]);

    // fi = (pos - min)/dx ; i = trunc(fi + 0.5)  (C-style trunc toward zero,
    // matching jnp astype(int32)). Keep the true fp32 division so boundary
    // rounding matches the reference bit-for-bit.
    float fx = (px - gmin) / dx + 0.5f;
    float fy = (py - gmin) / dx + 0.5f;
    float fz = (pz - gmin) / dx + 0.5f;
    int ix = (int)fx;
    int iy = (int)fy;
    int iz = (int)fz;

    bool ok = (ix >= 0) & (ix < GRID_N) &
              (iy >= 0) & (iy < GRID_N) &
              (iz >= 0) & (iz < GRID_N);
    if (ok) {
      int flat = (ix * GRID_N + iy) * GRID_N + iz;
      // Hardware global_atomic_add_f32, no return value needed.
      unsafeAtomicAdd(&grid[flat], w);
    }
  }
}

__global__ void __launch_bounds__(256) zero_grid_kernel(float4* __restrict__ g,
                                                        int n4) {
  int i = blockIdx.x * blockDim.x + threadIdx.x;
  if (i < n4) {
    g[i] = make_float4(0.0f, 0.0f, 0.0f, 0.0f);
  }
}

extern "C" void kernel_launch(void* const* d_in, const int* in_sizes, int n_in,
                              void* d_out, int out_size, void* d_ws, size_t ws_size,
                              hipStream_t stream) {
  const float* positions = (const float*)d_in[0];  // N*3 floats
  const float* weights   = (const float*)d_in[1];  // N floats
  float* grid            = (float*)d_out;          // 256^3 floats

  const int n = in_sizes[1];  // number of particles

  // 1) Zero the output grid (harness poisons d_out). 128-bit stores.
  {
    const int n4 = out_size / 4;
    const int threads = 256;
    const int blocks = (n4 + threads - 1) / threads;
    zero_grid_kernel<<<blocks, threads, 0, stream>>>((float4*)grid, n4);
  }

  // 2) Scatter with grid-stride loop: ~8 particles per thread keeps enough
  //    waves in flight to saturate L2 atomic throughput without launch bloat.
  {
    const int threads = 256;  // 8 wave32 per block
    const int per_thread = 8;
    int blocks = (n + threads * per_thread - 1) / (threads * per_thread);
    if (blocks < 1) blocks = 1;
    scatter_kernel<<<blocks, threads, 0, stream>>>(positions, weights, grid, n);
  }
}